// spuLayerTransformer_79259326480985
// MI455X (gfx1250) — compile-verified
//
#include <hip/hip_runtime.h>

// Elementwise SPU + box-bounds kernel for MI455X (gfx1250, wave32).
// Bandwidth-bound: 384 MB streamed @ 23.3 TB/s -> ~16.5 us floor.
// Strategy: float4 (b128) non-temporal loads/stores, branch-free select math.

typedef __attribute__((ext_vector_type(4))) float v4f;

__device__ __forceinline__ float spu_scalar(float z) {
    // SPU(z) = z^2 - 0.5            for z >= 0
    //        = sigmoid(-z) - 1      for z <  0
    // sigmoid(-z) - 1 = -e^z / (1 + e^z); for z < 0, e^z in (0,1): safe.
    float pos = __builtin_fmaf(z, z, -0.5f);
    float e   = __expf(z);                 // v_exp_f32 (TRANS); NaN/inf only in dead lane-values
    float neg = -e / (1.0f + e);
    return (z >= 0.0f) ? pos : neg;
}

__device__ __forceinline__ v4f spu_v4(v4f z) {
    v4f r;
    r.x = spu_scalar(z.x);
    r.y = spu_scalar(z.y);
    r.z = spu_scalar(z.z);
    r.w = spu_scalar(z.w);
    return r;
}

__global__ __launch_bounds__(256) void spu_box_kernel(
    const float* __restrict__ x,
    const float* __restrict__ l,
    const float* __restrict__ u,
    float* __restrict__ x_out,
    float* __restrict__ l_out,
    float* __restrict__ u_out,
    int n)
{
    const long long base = ((long long)blockIdx.x * blockDim.x + threadIdx.x) * 4;

    if (base + 3 < (long long)n) {
        // Streaming path: 128-bit non-temporal loads/stores (global_*_b128, TH=NT).
        v4f xv = __builtin_nontemporal_load((const v4f*)(x + base));
        v4f lv = __builtin_nontemporal_load((const v4f*)(l + base));
        v4f uv = __builtin_nontemporal_load((const v4f*)(u + base));

        v4f sx = spu_v4(xv);
        v4f sl = spu_v4(lv);
        v4f su = spu_v4(uv);

        v4f lo, uo;
        // l_out = l>=0 ? spu(l) : (u<=0 ? spu(u) : -0.5)
        // u_out = u<=0 ? spu(l) : spu(u)
        lo.x = (lv.x >= 0.0f) ? sl.x : ((uv.x <= 0.0f) ? su.x : -0.5f);
        lo.y = (lv.y >= 0.0f) ? sl.y : ((uv.y <= 0.0f) ? su.y : -0.5f);
        lo.z = (lv.z >= 0.0f) ? sl.z : ((uv.z <= 0.0f) ? su.z : -0.5f);
        lo.w = (lv.w >= 0.0f) ? sl.w : ((uv.w <= 0.0f) ? su.w : -0.5f);

        uo.x = (uv.x <= 0.0f) ? sl.x : su.x;
        uo.y = (uv.y <= 0.0f) ? sl.y : su.y;
        uo.z = (uv.z <= 0.0f) ? sl.z : su.z;
        uo.w = (uv.w <= 0.0f) ? sl.w : su.w;

        __builtin_nontemporal_store(sx, (v4f*)(x_out + base));
        __builtin_nontemporal_store(lo, (v4f*)(l_out + base));
        __builtin_nontemporal_store(uo, (v4f*)(u_out + base));
    } else if (base < (long long)n) {
        // Scalar tail (n may not be a multiple of 4; DIM is, but stay robust).
        for (long long i = base; i < (long long)n; ++i) {
            float lv = l[i];
            float uv = u[i];
            float sx = spu_scalar(x[i]);
            float sl = spu_scalar(lv);
            float su = spu_scalar(uv);
            x_out[i] = sx;
            l_out[i] = (lv >= 0.0f) ? sl : ((uv <= 0.0f) ? su : -0.5f);
            u_out[i] = (uv <= 0.0f) ? sl : su;
        }
    }
}

extern "C" void kernel_launch(void* const* d_in, const int* in_sizes, int n_in,
                              void* d_out, int out_size, void* d_ws, size_t ws_size,
                              hipStream_t stream)
{
    (void)n_in; (void)d_ws; (void)ws_size; (void)out_size;

    const float* x = (const float*)d_in[0];
    const float* l = (const float*)d_in[1];
    const float* u = (const float*)d_in[2];
    const int n = in_sizes[0];

    float* out   = (float*)d_out;
    float* x_out = out;
    float* l_out = out + (size_t)n;
    float* u_out = out + 2 * (size_t)n;

    const int threads = 256;                       // 8 wave32 waves per block
    const long long nvec = ((long long)n + 3) / 4; // one float4 per thread
    const int blocks = (int)((nvec + threads - 1) / threads);

    spu_box_kernel<<<blocks, threads, 0, stream>>>(x, l, u, x_out, l_out, u_out, n);
}